// ModelPart21_90305982366372
// MI455X (gfx1250) — compile-verified
//
#include <hip/hip_runtime.h>
#include <math.h>

#define KPOOL 7
#define NCELL 49
#define HH 256
#define WW 256
#define CCH 196          // 4 * 49 channels
#define NPIX (HH * WW)
#define PIX_PER_BLOCK 32
#define CHUNKS_PER_BLOCK ((PIX_PER_BLOCK * CCH) / 4)   // 1568 x 16B = 25088 B

// ---------------------------------------------------------------------------
// gfx1250 async global->LDS helpers (ASYNCcnt-tracked, no VGPR round trip)
// ---------------------------------------------------------------------------
__device__ __forceinline__ void async_load_b128(unsigned lds_byte_off, const void* gaddr) {
    asm volatile("global_load_async_to_lds_b128 %0, %1, off"
                 :: "v"(lds_byte_off), "v"(gaddr)
                 : "memory");
}
__device__ __forceinline__ void wait_asynccnt0() {
    asm volatile("s_wait_asynccnt 0x0" ::: "memory");
}

// ---------------------------------------------------------------------------
// Repack (H,W,196) -> 49 planes of (H,W,4): rep[cell][y*W+x] = {d0,d1,d2,d3}
// where original channel = d*49 + cell.
// Stage 1: coalesced async b128 copies of 32 pixels' channel vectors to LDS.
// Stage 2: strided LDS reads -> coalesced float4 global stores per cell plane.
// ---------------------------------------------------------------------------
__global__ void repack_kernel(const float* __restrict__ in, float4* __restrict__ out) {
    __shared__ __align__(16) float lds[PIX_PER_BLOCK * CCH];   // 25088 B

    const int tid  = threadIdx.x;
    const int pix0 = blockIdx.x * PIX_PER_BLOCK;

    const char* gbase = (const char*)(in + (size_t)pix0 * CCH);
    const unsigned lds_base = (unsigned)(uintptr_t)(&lds[0]);  // low 32 bits == LDS offset

    // Stage 1: 1568 16B chunks, consecutive lanes -> consecutive chunks
    for (int c = tid; c < CHUNKS_PER_BLOCK; c += blockDim.x) {
        async_load_b128(lds_base + (unsigned)(c * 16), gbase + (size_t)c * 16);
    }
    wait_asynccnt0();
    __syncthreads();

    // Stage 2: 1568 float4 outputs (49 cells x 32 pixels)
    for (int o = tid; o < NCELL * PIX_PER_BLOCK; o += blockDim.x) {
        const int cell = o >> 5;          // [0,49)
        const int p    = o & 31;          // [0,32)
        const float* row = &lds[p * CCH + cell];
        float4 v;
        v.x = row[0];
        v.y = row[49];
        v.z = row[98];
        v.w = row[147];
        out[((size_t)cell << 16) + pix0 + p] = v;
    }
}

__device__ __forceinline__ void cell_coords(int cell, float x1, float y1, float rw, float rh,
                                            int& y0, int& x0, int& y1i, int& x1i,
                                            float& wy, float& wx) {
    const int i = cell / KPOOL;
    const int j = cell - i * KPOOL;
    const float inv7 = 1.0f / 7.0f;
    float cy = fminf(fmaxf(fmaf((i + 0.5f) * inv7, rh, y1), 0.0f), (float)(HH - 1));
    float cx = fminf(fmaxf(fmaf((j + 0.5f) * inv7, rw, x1), 0.0f), (float)(WW - 1));
    float y0f = floorf(cy);
    float x0f = floorf(cx);
    wy = cy - y0f;
    wx = cx - x0f;
    y0 = (int)y0f;
    x0 = (int)x0f;
    y1i = min(y0 + 1, HH - 1);
    x1i = min(x0 + 1, WW - 1);
}

__device__ __forceinline__ float4 bilin_rep(const float4* __restrict__ sm, int cell,
                                            int y0, int x0, int y1i, int x1i,
                                            float wy, float wx) {
    const float4* base = sm + ((size_t)cell << 16);
    float4 g00 = base[y0  * WW + x0 ];
    float4 g01 = base[y0  * WW + x1i];
    float4 g10 = base[y1i * WW + x0 ];
    float4 g11 = base[y1i * WW + x1i];
    float w00 = (1.0f - wy) * (1.0f - wx);
    float w01 = (1.0f - wy) * wx;
    float w10 = wy * (1.0f - wx);
    float w11 = wy * wx;
    float4 r;
    r.x = w00 * g00.x + w01 * g01.x + w10 * g10.x + w11 * g11.x;
    r.y = w00 * g00.y + w01 * g01.y + w10 * g10.y + w11 * g11.y;
    r.z = w00 * g00.z + w01 * g01.z + w10 * g10.z + w11 * g11.z;
    r.w = w00 * g00.w + w01 * g01.w + w10 * g10.w + w11 * g11.w;
    return r;
}

__device__ __forceinline__ float4 bilin_orig(const float* __restrict__ sm, int cell,
                                             int y0, int x0, int y1i, int x1i,
                                             float wy, float wx) {
    const float* p00 = sm + (size_t)(y0  * WW + x0 ) * CCH + cell;
    const float* p01 = sm + (size_t)(y0  * WW + x1i) * CCH + cell;
    const float* p10 = sm + (size_t)(y1i * WW + x0 ) * CCH + cell;
    const float* p11 = sm + (size_t)(y1i * WW + x1i) * CCH + cell;
    float w00 = (1.0f - wy) * (1.0f - wx);
    float w01 = (1.0f - wy) * wx;
    float w10 = wy * (1.0f - wx);
    float w11 = wy * wx;
    float4 r;
    r.x = w00 * p00[0]   + w01 * p01[0]   + w10 * p10[0]   + w11 * p11[0];
    r.y = w00 * p00[49]  + w01 * p01[49]  + w10 * p10[49]  + w11 * p11[49];
    r.z = w00 * p00[98]  + w01 * p01[98]  + w10 * p10[98]  + w11 * p11[98];
    r.w = w00 * p00[147] + w01 * p01[147] + w10 * p10[147] + w11 * p11[147];
    return r;
}

// ---------------------------------------------------------------------------
// One wave32 per ROI. Lane l handles cell l (always) and cell l+32 (l < 17).
// Cross-lane reduce two class sums, uniform 2-way softmax head, then each
// lane writes its cells' mask float2.
// ---------------------------------------------------------------------------
template <bool REPACKED>
__global__ void psroi_head_kernel(const float* __restrict__ roi,
                                  const float* __restrict__ smap,
                                  const float4* __restrict__ smrep,
                                  float* __restrict__ out, int N) {
    const int lane = threadIdx.x & 31;
    const int n = blockIdx.x * (blockDim.x >> 5) + (threadIdx.x >> 5);
    if (n >= N) return;   // whole wave exits together

    const float x1 = roi[4 * n + 0] * 0.125f;
    const float y1 = roi[4 * n + 1] * 0.125f;
    const float x2 = roi[4 * n + 2] * 0.125f;
    const float y2 = roi[4 * n + 3] * 0.125f;
    const float rw = fmaxf(x2 - x1, 1.0f);
    const float rh = fmaxf(y2 - y1, 1.0f);

    const int cellA = lane;
    const int cellB = lane + 32;
    const bool hasB = (lane < NCELL - 32);   // lanes 0..16

    int ya0, xa0, ya1, xa1; float wya, wxa;
    cell_coords(cellA, x1, y1, rw, rh, ya0, xa0, ya1, xa1, wya, wxa);
    int yb0 = 0, xb0 = 0, yb1 = 0, xb1 = 0; float wyb = 0.f, wxb = 0.f;
    if (hasB) cell_coords(cellB, x1, y1, rw, rh, yb0, xb0, yb1, xb1, wyb, wxb);

    if (REPACKED && hasB) {
        // gfx1250 global_prefetch_b8: warm cell B's lines while A's loads fly
        const float4* b2 = smrep + ((size_t)cellB << 16);
        __builtin_prefetch(&b2[yb0 * WW + xb0], 0, 3);
        __builtin_prefetch(&b2[yb0 * WW + xb1], 0, 3);
        __builtin_prefetch(&b2[yb1 * WW + xb0], 0, 3);
        __builtin_prefetch(&b2[yb1 * WW + xb1], 0, 3);
    }

    float4 va = REPACKED ? bilin_rep(smrep, cellA, ya0, xa0, ya1, xa1, wya, wxa)
                         : bilin_orig(smap,  cellA, ya0, xa0, ya1, xa1, wya, wxa);
    float sum0 = fmaxf(va.x, va.z);   // s=0 pairs with d in {0,2}
    float sum1 = fmaxf(va.y, va.w);   // s=1 pairs with d in {1,3}

    float4 vb = make_float4(0.f, 0.f, 0.f, 0.f);
    if (hasB) {
        vb = REPACKED ? bilin_rep(smrep, cellB, yb0, xb0, yb1, xb1, wyb, wxb)
                      : bilin_orig(smap,  cellB, yb0, xb0, yb1, xb1, wyb, wxb);
        sum0 += fmaxf(vb.x, vb.z);
        sum1 += fmaxf(vb.y, vb.w);
    }

    // wave32 butterfly reduction -> every lane holds the full sums
    #pragma unroll
    for (int off = 16; off >= 1; off >>= 1) {
        sum0 += __shfl_xor(sum0, off, 32);
        sum1 += __shfl_xor(sum1, off, 32);
    }

    const float ave0 = sum0 * (1.0f / 49.0f);
    const float ave1 = sum1 * (1.0f / 49.0f);
    const float mx = fmaxf(ave0, ave1);
    const float e0 = __expf(ave0 - mx);
    const float e1 = __expf(ave1 - mx);
    const float inv = 1.0f / (e0 + e1);
    const float p0 = e0 * inv;
    const float p1 = e1 * inv;
    const int   sel   = (ave1 > ave0) ? 1 : 0;   // jnp.argmax: first max wins
    const float score = fmaxf(p0, p1);

    if (lane == 0) {
        out[2 * n]              = p0;
        out[2 * n + 1]          = p1;
        out[(size_t)2 * N + n]  = (float)sel;    // cls_result
        out[(size_t)3 * N + n]  = score;         // cls_score
    }

    // mask_result[n, i, j, c] = softmax_c(v[c*2+sel]); cell index = i*7+j
    float* mout = out + (size_t)4 * N + (size_t)n * (2 * NCELL);
    {
        float a = sel ? va.y : va.x;
        float b = sel ? va.w : va.z;
        float m = fmaxf(a, b);
        float ea = __expf(a - m), eb = __expf(b - m);
        float r = 1.0f / (ea + eb);
        *(float2*)(mout + cellA * 2) = make_float2(ea * r, eb * r);
    }
    if (hasB) {
        float a = sel ? vb.y : vb.x;
        float b = sel ? vb.w : vb.z;
        float m = fmaxf(a, b);
        float ea = __expf(a - m), eb = __expf(b - m);
        float r = 1.0f / (ea + eb);
        *(float2*)(mout + cellB * 2) = make_float2(ea * r, eb * r);
    }
}

extern "C" void kernel_launch(void* const* d_in, const int* in_sizes, int n_in,
                              void* d_out, int out_size, void* d_ws, size_t ws_size,
                              hipStream_t stream) {
    (void)n_in; (void)out_size;
    const float* roi  = (const float*)d_in[0];   // (N,4) f32
    const float* smap = (const float*)d_in[1];   // (1,256,256,196) f32
    float* out = (float*)d_out;
    const int N = in_sizes[0] / 4;

    const int rois_per_block = 256 / 32;                       // 8 waves/block
    const int blocks = (N + rois_per_block - 1) / rois_per_block;

    const size_t repack_bytes = (size_t)NCELL * NPIX * 4 * sizeof(float);  // ~51.4 MB
    if (d_ws != nullptr && ws_size >= repack_bytes) {
        float4* rep = (float4*)d_ws;
        repack_kernel<<<NPIX / PIX_PER_BLOCK, 256, 0, stream>>>(smap, rep);
        psroi_head_kernel<true><<<blocks, 256, 0, stream>>>(roi, smap, rep, out, N);
    } else {
        psroi_head_kernel<false><<<blocks, 256, 0, stream>>>(roi, smap, nullptr, out, N);
    }
}